// EEGGCN_19069654794648
// MI455X (gfx1250) — compile-verified
//
#include <hip/hip_runtime.h>
#include <hip/hip_bf16.h>

typedef __attribute__((ext_vector_type(2))) float v2f;
typedef __attribute__((ext_vector_type(8))) float v8f;

#if __has_builtin(__builtin_amdgcn_global_load_async_to_lds_b32)
#define GCN_HAVE_ASYNC_LDS 1
typedef int __attribute__((address_space(1)))* gcn_gptr_t;
typedef int __attribute__((address_space(3)))* gcn_lptr_t;
#endif

__device__ __forceinline__ void gcn_wait_async_then_barrier() {
#ifdef GCN_HAVE_ASYNC_LDS
#if __has_builtin(__builtin_amdgcn_s_wait_asynccnt)
    __builtin_amdgcn_s_wait_asynccnt(0);
#else
    asm volatile("s_wait_asynccnt 0x0" ::: "memory");
#endif
#endif
    __syncthreads();
}

// ---------------------------------------------------------------------------
// Degree accumulation: deg[dst[e]] += 1 (self loop folded in as +1 later).
// ---------------------------------------------------------------------------
__global__ __launch_bounds__(256) void gcn_deg_kernel(const int* __restrict__ dst,
                                                      float* __restrict__ deg,
                                                      int nEdges) {
    int e = blockIdx.x * blockDim.x + threadIdx.x;
    if (e < nEdges) {
        atomicAdd(&deg[dst[e]], 1.0f);
    }
}

// dinv[i] = rsqrt(deg[i] + 1)   (+1 = self loop; always > 0)
__global__ __launch_bounds__(256) void gcn_dinv_kernel(float* __restrict__ deg, int n) {
    int i = blockIdx.x * blockDim.x + threadIdx.x;
    if (i < n) {
        deg[i] = rsqrtf(deg[i] + 1.0f);
    }
}

// ---------------------------------------------------------------------------
// Node-feature GEMM via V_WMMA_F32_16X16X4_F32, LDS-staged operands.
//   out[n x Hout] = A[n x K] @ W[K x Hout]      (row-major fp32, K,Hout pow2)
// Block: 256 threads = 8 waves; block tile = 32 rows x 64 cols;
// wave tile = 16x16 (mtile = wave>>2, ntile = wave&3).
// LDS: A tile [32][KP] and W transposed [64][KP], KP = K+4 so the wave32
// fragment reads (lane strides of 4 dwords mod 64 banks) are conflict-free.
// Inner loop: 2x ds_load_b64 + v_wmma_f32_16x16x4_f32, no divergence.
// ---------------------------------------------------------------------------
#define GCN_MAX_KP (128 + 4)

__global__ __launch_bounds__(256) void gcn_gemm_wmma_kernel(
        const float* __restrict__ A, const float* __restrict__ W,
        float* __restrict__ out, int n, int K, int Hout, int Ksh, int Hsh) {
    __shared__ float ldsA[32 * GCN_MAX_KP];
    __shared__ float ldsW[64 * GCN_MAX_KP];

    const int KP      = K + 4;
    const int tid     = threadIdx.x;
    const int row0blk = blockIdx.x * 32;

    // Stage W transposed: ldsW[c*KP + k] = W[k*Hout + c].  Coalesced global
    // reads; on gfx1250 use the async-to-LDS path (tracked by ASYNCcnt).
    for (int idx = tid; idx < (K << Hsh); idx += 256) {
        const int k = idx >> Hsh;
        const int c = idx & (Hout - 1);
#ifdef GCN_HAVE_ASYNC_LDS
        __builtin_amdgcn_global_load_async_to_lds_b32(
            (gcn_gptr_t)(W + idx), (gcn_lptr_t)&ldsW[c * KP + k], 0, 0);
#else
        ldsW[c * KP + k] = W[idx];
#endif
    }

    // Stage A tile with branch-free OOB padding (select, no exec divergence).
    for (int idx = tid; idx < (K << 5); idx += 256) {
        const int r    = idx >> Ksh;
        const int k    = idx & (K - 1);
        const int row  = row0blk + r;
        const int rowc = row < n ? row : (n - 1);
        const float v  = A[((size_t)rowc << Ksh) + k];
        ldsA[r * KP + k] = (row < n) ? v : 0.0f;
    }

    gcn_wait_async_then_barrier();

    const int lane  = tid & 31;
    const int wave  = tid >> 5;
    const int mtile = wave >> 2;             // 0..1
    const int ntile = wave & 3;              // 0..3
    const int lm    = lane & 15;             // M (A) / N (B, store) index
    const int khalf = (lane >> 4) << 1;      // 0 | 2

    const float* aP = &ldsA[(mtile * 16 + lm) * KP + khalf];
    const float* bP = &ldsW[(ntile * 16 + lm) * KP + khalf];

    v8f acc = {};
    #pragma unroll 4
    for (int k0 = 0; k0 < K; k0 += 4) {
        const v2f a = *(const v2f*)(aP + k0);
        const v2f b = *(const v2f*)(bP + k0);
        acc = __builtin_amdgcn_wmma_f32_16x16x4_f32(
            /*neg_a=*/false, a, /*neg_b=*/false, b,
            /*c_mod=*/(short)0, acc, /*reuse_a=*/false, /*reuse_b=*/false);
    }

    // C/D layout: VGPR v -> M = v (lanes 0-15) | v+8 (lanes 16-31), N = lane%16
    const int rbase = row0blk + mtile * 16 + ((lane >> 4) << 3);
    const int col   = ntile * 16 + lm;
    float* op = out + (((size_t)rbase) << Hsh) + col;
    if (rbase + 7 < n) {
        #pragma unroll
        for (int v = 0; v < 8; ++v) {
            op[(size_t)v << Hsh] = acc[v];
        }
    } else {
        for (int v = 0; v < 8; ++v) {
            if (rbase + v < n) op[(size_t)v << Hsh] = acc[v];
        }
    }
}

// ---------------------------------------------------------------------------
// Edge scatter: one wave per edge; each lane handles 2 consecutive features
// via one b64 gather.  out[dst] += h[src] * dinv[src]*dinv[dst].
// atomic f32 adds resolve in L2 (25.6MB tables << 192MB L2).
// ---------------------------------------------------------------------------
__global__ __launch_bounds__(256) void gcn_edge_scatter_kernel(
        const float* __restrict__ h, const int* __restrict__ src,
        const int* __restrict__ dst, const float* __restrict__ dinv,
        float* __restrict__ out, int nEdges, int H, int Hsh) {
    const int e = blockIdx.x * 8 + (threadIdx.x >> 5);
    if (e >= nEdges) return;
    const int lane = threadIdx.x & 31;
    const int s = src[e];
    const int d = dst[e];
    const float norm = dinv[s] * dinv[d];
    const float2* hs = (const float2*)(h + ((size_t)s << Hsh));
    float* od = out + ((size_t)d << Hsh);
    for (int f2 = lane; (f2 << 1) < H; f2 += 32) {   // one iteration for H=64
        const float2 hv = hs[f2];
        atomicAdd(&od[2 * f2 + 0], hv.x * norm);
        atomicAdd(&od[2 * f2 + 1], hv.y * norm);
    }
}

// ---------------------------------------------------------------------------
// Epilogue: out[i,f] = relu(out[i,f] + h[i,f]*dinv[i]^2 + bias[f])
// (adds the self-loop contribution and bias, then ReLU, in place)
// ---------------------------------------------------------------------------
__global__ __launch_bounds__(256) void gcn_finish_relu_kernel(
        float* __restrict__ out, const float* __restrict__ h,
        const float* __restrict__ dinv, const float* __restrict__ bias,
        int total, int H, int Hsh) {
    const int i = blockIdx.x * blockDim.x + threadIdx.x;
    if (i >= total) return;
    const int node = i >> Hsh;
    const int f    = i & (H - 1);
    const float di = dinv[node];
    const float v  = out[i] + h[i] * di * di + bias[f];
    out[i] = v > 0.0f ? v : 0.0f;
}

// ---------------------------------------------------------------------------
// Segment-sum pooling: pooled[batch[node]] += h[node]; cnt[batch[node]] += 1
// ---------------------------------------------------------------------------
__global__ __launch_bounds__(256) void gcn_pool_kernel(
        const float* __restrict__ h, const int* __restrict__ batch,
        float* __restrict__ pooled, float* __restrict__ cnt,
        int total, int H, int Hsh) {
    const int i = blockIdx.x * blockDim.x + threadIdx.x;
    if (i >= total) return;
    const int node = i >> Hsh;
    const int f    = i & (H - 1);
    const int g    = batch[node];
    atomicAdd(&pooled[((size_t)g << Hsh) + f], h[i]);
    if (f == 0) {
        atomicAdd(&cnt[g], 1.0f);
    }
}

// Final head: out[b,c] = (pooled[b]/max(cnt,1)) . Wl[:,c] + bl[c]
__global__ __launch_bounds__(256) void gcn_final_kernel(
        const float* __restrict__ pooled, const float* __restrict__ cnt,
        const float* __restrict__ Wl, const float* __restrict__ bl,
        float* __restrict__ out, int B, int H, int C) {
    const int i = blockIdx.x * blockDim.x + threadIdx.x;
    if (i >= B * C) return;
    const int b = i / C;
    const int c = i - b * C;
    const float inv = 1.0f / fmaxf(cnt[b], 1.0f);
    float acc = bl[c];
    for (int k = 0; k < H; ++k) {
        acc += pooled[(size_t)b * H + k] * inv * Wl[(size_t)k * C + c];
    }
    out[i] = acc;
}

// ---------------------------------------------------------------------------
extern "C" void kernel_launch(void* const* d_in, const int* in_sizes, int n_in,
                              void* d_out, int out_size, void* d_ws, size_t ws_size,
                              hipStream_t stream) {
    const float* x   = (const float*)d_in[0];
    const int*   ei  = (const int*)d_in[1];   // edge_index [2, E]
    const int*   bat = (const int*)d_in[2];   // batch [N]
    const float* W1  = (const float*)d_in[3];
    const float* b1  = (const float*)d_in[4];
    const float* W2  = (const float*)d_in[5];
    const float* b2  = (const float*)d_in[6];
    const float* Wl  = (const float*)d_in[7];
    const float* bl  = (const float*)d_in[8];
    float* out = (float*)d_out;

    const int N    = in_sizes[2];
    const int E    = in_sizes[1] / 2;
    const int F_IN = in_sizes[0] / N;              // 128
    const int H    = in_sizes[4];                  // 64
    const int C    = in_sizes[8];                  // 2
    const int B    = out_size / C;                 // 256
    const int Fsh  = __builtin_ctz(F_IN);          // 7
    const int Hsh  = __builtin_ctz(H);             // 6

    const int* src = ei;
    const int* dst = ei + E;

    // Workspace layout (floats)
    float* ws     = (float*)d_ws;
    float* dinv   = ws;                            // N (deg -> rsqrt in place)
    float* buf0   = dinv + N;                      // N*H (h1, reused as h2)
    float* buf1   = buf0 + (size_t)N * H;          // N*H (out1/relu1, reused as out2)
    float* pooled = buf1 + (size_t)N * H;          // B*H
    float* cnt    = pooled + (size_t)B * H;        // B

    const dim3 blk(256);
    const dim3 gemmGrid((N + 31) / 32);
    const dim3 edgeGrid((E + 7) / 8);
    const int  totalNH = N * H;
    const dim3 nhGrid((totalNH + 255) / 256);

    // --- degrees & normalization ---
    (void)hipMemsetAsync(dinv, 0, (size_t)N * sizeof(float), stream);
    gcn_deg_kernel<<<(E + 255) / 256, blk, 0, stream>>>(dst, dinv, E);
    gcn_dinv_kernel<<<(N + 255) / 256, blk, 0, stream>>>(dinv, N);

    // --- layer 1 ---
    gcn_gemm_wmma_kernel<<<gemmGrid, blk, 0, stream>>>(x, W1, buf0, N, F_IN, H, Fsh, Hsh);
    (void)hipMemsetAsync(buf1, 0, (size_t)N * H * sizeof(float), stream);
    gcn_edge_scatter_kernel<<<edgeGrid, blk, 0, stream>>>(buf0, src, dst, dinv, buf1, E, H, Hsh);
    gcn_finish_relu_kernel<<<nhGrid, blk, 0, stream>>>(buf1, buf0, dinv, b1, totalNH, H, Hsh);

    // --- layer 2 ---
    gcn_gemm_wmma_kernel<<<gemmGrid, blk, 0, stream>>>(buf1, W2, buf0, N, H, H, Hsh, Hsh);
    (void)hipMemsetAsync(buf1, 0, (size_t)N * H * sizeof(float), stream);
    gcn_edge_scatter_kernel<<<edgeGrid, blk, 0, stream>>>(buf0, src, dst, dinv, buf1, E, H, Hsh);
    gcn_finish_relu_kernel<<<nhGrid, blk, 0, stream>>>(buf1, buf0, dinv, b2, totalNH, H, Hsh);

    // --- pooling + head ---
    (void)hipMemsetAsync(pooled, 0, (size_t)(B * H + B) * sizeof(float), stream);
    gcn_pool_kernel<<<nhGrid, blk, 0, stream>>>(buf1, bat, pooled, cnt, totalNH, H, Hsh);
    gcn_final_kernel<<<(B * C + 255) / 256, blk, 0, stream>>>(pooled, cnt, Wl, bl, out, B, H, C);
}